// Mixture_Loss_74053826118054
// MI455X (gfx1250) — compile-verified
//
#include <hip/hip_runtime.h>

// ---------------------------------------------------------------------------
// Mixture loss for (S=64, B=256, D=1024) f32 tensors on gfx1250.
// Pass 1: per-(b, 16-row s-chunk, K-half) Gram matrices XX^T, TT^T, XT^T via
//         V_WMMA_F32_16X16X4_F32, LDS-staged tiles. Diagonals give ||l||^2,
//         ||t||^2, dot(l,t); superdiagonals give adjacent-step dots.
// Pass 2: single block combines, masks, compacts valid pairs (scan), dd term.
// ---------------------------------------------------------------------------

typedef float v2f __attribute__((ext_vector_type(2)));
typedef float v8f __attribute__((ext_vector_type(8)));

#define S_DIM 64
#define B_DIM 256
#define D_DIM 1024
#define BS (B_DIM * S_DIM)          // 16384
#define ROWSTRIDE (B_DIM * D_DIM)   // element stride between consecutive s
#define CH 16                       // s-rows per chunk (WMMA M=N=16)
#define KT 64                       // K-tile staged in LDS
#define LDSW 68                     // padded LDS row stride (floats)
#define NPAIR (B_DIM * (S_DIM - 1)) // 16128

// C/D 16x16 f32 layout: element (M,N) lives in VGPR (M&7), lane N+16*(M>>3).
// We dump lane l's 8 acc regs at lds[l*8 + r].
__device__ __forceinline__ int cidx(int M, int N) {
  return (N + 16 * (M >> 3)) * 8 + (M & 7);
}

__global__ void __launch_bounds__(32) gram_pass1(const float* __restrict__ logits,
                                                 const float* __restrict__ tgt,
                                                 float* __restrict__ ws) {
  __shared__ float ldsX[CH * LDSW];
  __shared__ float ldsT[CH * LDSW];

  const int lane  = threadIdx.x;
  const int bid   = blockIdx.x;          // 256*4*2 blocks of one wave each
  const int kh    = bid & 1;             // K half: [0,512) or [512,1024)
  const int chunk = (bid >> 1) & 3;      // s-chunk
  const int b     = bid >> 3;            // batch
  const int s0    = chunk * CH;
  const int kb0   = kh * (D_DIM / 2);
  const bool hasB = (s0 + CH) < S_DIM;   // boundary pair (s0+15, s0+16) exists

  const float* Xb = logits + ((size_t)s0 * B_DIM + b) * D_DIM;
  const float* Tb = tgt    + ((size_t)s0 * B_DIM + b) * D_DIM;

  const int m = lane & 15;   // A: row M / B: col N held by this lane
  const int h = lane >> 4;   // K half within fragment

  v8f xx = {};   // X * X^T
  v8f tts = {};  // T * T^T
  v8f xt = {};   // X * T^T
  float bl = 0.f, bt = 0.f;  // boundary adjacent dots (rows 15,16)

  for (int t = 0; t < (D_DIM / 2) / KT; ++t) {
    const int kb = kb0 + t * KT;
    // ---- stage 16x64 tiles of X and T into LDS (coalesced b128 loads) ----
#pragma unroll
    for (int j = 0; j < 8; ++j) {
      const int e  = j * 32 + lane;
      const int r  = e >> 4;           // tile row 0..15
      const int c4 = (e & 15) << 2;    // col 0..60 step 4
      *(float4*)&ldsX[r * LDSW + c4] =
          *(const float4*)(Xb + (size_t)r * ROWSTRIDE + kb + c4);
      *(float4*)&ldsT[r * LDSW + c4] =
          *(const float4*)(Tb + (size_t)r * ROWSTRIDE + kb + c4);
    }
    // prefetch next tile into cache hierarchy (global_prefetch_b8)
    if (t + 1 < (D_DIM / 2) / KT) {
      const int kn = kb + KT;
#pragma unroll
      for (int j = 0; j < 8; ++j) {
        const int e  = j * 32 + lane;
        const int r  = e >> 4;
        const int c4 = (e & 15) << 2;
        __builtin_prefetch(Xb + (size_t)r * ROWSTRIDE + kn + c4, 0, 1);
        __builtin_prefetch(Tb + (size_t)r * ROWSTRIDE + kn + c4, 0, 1);
      }
    }
    __syncthreads();

    // ---- boundary adjacent-row dot: LDS row 15 x global row 16 (VALU) ----
    if (hasB) {
      const float2 gx = *(const float2*)(Xb + (size_t)CH * ROWSTRIDE + kb + lane * 2);
      const float2 gt = *(const float2*)(Tb + (size_t)CH * ROWSTRIDE + kb + lane * 2);
      const float2 lx = *(const float2*)&ldsX[15 * LDSW + lane * 2];
      const float2 lz = *(const float2*)&ldsT[15 * LDSW + lane * 2];
      bl += lx.x * gx.x + lx.y * gx.y;
      bt += lz.x * gt.x + lz.y * gt.y;
    }

    // ---- 16 K-steps of 4: three Grams per step ----
    // A frag 16x4 f32: lane m, VGPR0=K(2h), VGPR1=K(2h+1)  (ISA 7.12.2).
    // B frag of the SAME matrix transposed has identical per-lane contents,
    // so one fragment per tensor feeds all three WMMAs.
#pragma unroll
    for (int kk = 0; kk < KT; kk += 4) {
      const v2f ax = *(const v2f*)&ldsX[m * LDSW + kk + 2 * h];
      const v2f av = *(const v2f*)&ldsT[m * LDSW + kk + 2 * h];
      xx  = __builtin_amdgcn_wmma_f32_16x16x4_f32(false, ax, false, ax, (short)0, xx,  false, false);
      tts = __builtin_amdgcn_wmma_f32_16x16x4_f32(false, av, false, av, (short)0, tts, false, false);
      xt  = __builtin_amdgcn_wmma_f32_16x16x4_f32(false, ax, false, av, (short)0, xt,  false, false);
    }
    __syncthreads();
  }

  // ---- dump accumulators to LDS for easy (M,N) extraction ----
  float* sXX = ldsX;        // 256 floats
  float* sXT = ldsX + 256;  // 256 floats
  float* sTT = ldsT;        // 256 floats
#pragma unroll
  for (int r = 0; r < 8; ++r) {
    sXX[lane * 8 + r] = xx[r];
    sTT[lane * 8 + r] = tts[r];
    sXT[lane * 8 + r] = xt[r];
  }
  __syncthreads();

  // ---- reduce boundary dots across the wave ----
  if (hasB) {
    for (int off = 16; off > 0; off >>= 1) {
      bl += __shfl_xor(bl, off, 32);
      bt += __shfl_xor(bt, off, 32);
    }
  }

  // ---- write partial results (per K-half buffers -> deterministic) ----
  float* NL = ws;            // [2][B][S]
  float* NT = ws + 2 * BS;
  float* LT = ws + 4 * BS;
  float* DL = ws + 6 * BS;
  float* DT = ws + 8 * BS;
  const int obase = kh * BS + b * S_DIM + s0;
  if (lane < 16) {
    NL[obase + m] = sXX[cidx(m, m)];
    NT[obase + m] = sTT[cidx(m, m)];
    LT[obase + m] = sXT[cidx(m, m)];
    if (m < 15) {
      DL[obase + m] = sXX[cidx(m, m + 1)];
      DT[obase + m] = sTT[cidx(m, m + 1)];
    }
  }
  if (hasB && lane == 0) {
    DL[obase + 15] = bl;
    DT[obase + 15] = bt;
  }
}

__device__ __forceinline__ float blockReduceF(float v, float* red) {
  const int t = threadIdx.x;
  red[t] = v;
  __syncthreads();
  for (int st = 128; st > 0; st >>= 1) {
    if (t < st) red[t] += red[t + st];
    __syncthreads();
  }
  const float r = red[0];
  __syncthreads();
  return r;
}

__global__ void __launch_bounds__(256) finalize_pass2(const int* __restrict__ mask,
                                                      float* __restrict__ ws,
                                                      float* __restrict__ out) {
  __shared__ float red[256];
  __shared__ int cnts[256];
  __shared__ int offs[257];
  const int b = threadIdx.x;  // one thread per batch element

  const float* NL0 = ws;           const float* NL1 = ws + BS;
  const float* NT0 = ws + 2 * BS;  const float* NT1 = ws + 3 * BS;
  const float* LT0 = ws + 4 * BS;  const float* LT1 = ws + 5 * BS;
  const float* DL0 = ws + 6 * BS;  const float* DL1 = ws + 7 * BS;
  const float* DT0 = ws + 8 * BS;  const float* DT1 = ws + 9 * BS;
  float* DCL = ws + 10 * BS;
  float* DCT = ws + 10 * BS + NPAIR;

  float mseS = 0.f, cosS = 0.f, delS = 0.f;
  int vcnt = 0, pcnt = 0;
  float nl_p = 0.f, nt_p = 0.f;
  int v_p = 0;
  for (int s = 0; s < S_DIM; ++s) {
    const int i = b * S_DIM + s;
    const float nl = NL0[i] + NL1[i];
    const float nt = NT0[i] + NT1[i];
    const float lt = LT0[i] + LT1[i];
    const int v = (mask[i] == 0);
    vcnt += v;
    if (v) {
      mseS += nl + nt - 2.f * lt;  // sum((l-t)^2) for this row
      const float c = lt / (fmaxf(sqrtf(nl), 1e-8f) * fmaxf(sqrtf(nt), 1e-8f));
      cosS += 1.f - c;
    }
    if (s > 0 && (v_p & v)) {
      const float dl = (DL0[i - 1] + DL1[i - 1]) /
                       (fmaxf(sqrtf(nl_p), 1e-6f) * fmaxf(sqrtf(nl), 1e-6f));
      const float dt = (DT0[i - 1] + DT1[i - 1]) /
                       (fmaxf(sqrtf(nt_p), 1e-6f) * fmaxf(sqrtf(nt), 1e-6f));
      const float d = dl - dt;
      delS += d * d;
      pcnt++;
    }
    nl_p = nl; nt_p = nt; v_p = v;
  }

  // exclusive scan of per-b valid-pair counts (flatten order is b-major,
  // so each thread's compacted pairs are contiguous -> matches cumsum/scatter)
  cnts[b] = pcnt;
  __syncthreads();
  if (b == 0) {
    int acc = 0;
    for (int i = 0; i < 256; ++i) { offs[i] = acc; acc += cnts[i]; }
    offs[256] = acc;
  }
  __syncthreads();
  int k = offs[b];
  const int total = offs[256];

  // second sweep: write compacted d_l / d_t
  nl_p = 0.f; nt_p = 0.f; v_p = 0;
  for (int s = 0; s < S_DIM; ++s) {
    const int i = b * S_DIM + s;
    const float nl = NL0[i] + NL1[i];
    const float nt = NT0[i] + NT1[i];
    const int v = (mask[i] == 0);
    if (s > 0 && (v_p & v)) {
      DCL[k] = (DL0[i - 1] + DL1[i - 1]) /
               (fmaxf(sqrtf(nl_p), 1e-6f) * fmaxf(sqrtf(nl), 1e-6f));
      DCT[k] = (DT0[i - 1] + DT1[i - 1]) /
               (fmaxf(sqrtf(nt_p), 1e-6f) * fmaxf(sqrtf(nt), 1e-6f));
      ++k;
    }
    nl_p = nl; nt_p = nt; v_p = v;
  }
  __threadfence();
  __syncthreads();

  // dd term over compacted arrays: i in [0, total-1)
  float ddS = 0.f;
  for (int i = b; i < total - 1; i += 256) {
    const float a0 = DCL[i], a1 = DCL[i + 1];
    const float c0 = DCT[i], c1 = DCT[i + 1];
    const float ddl = (a1 - a0) / ((a0 != 0.f) ? a0 : 1e-6f);
    const float ddt = (c1 - c0) / ((c0 != 0.f) ? c0 : 1e-6f);
    const float d = ddl - ddt;
    ddS += d * d;
  }

  const float mseT = blockReduceF(mseS, red);
  const float cosT = blockReduceF(cosS, red);
  const float delT = blockReduceF(delS, red);
  const float ddT  = blockReduceF(ddS, red);
  const float nvT  = blockReduceF((float)vcnt, red);

  if (b == 0) {
    const float pc = (float)total;
    const float loss = mseT / (nvT * (float)D_DIM)
                     + cosT / nvT
                     + delT / fmaxf(pc, 1.f)
                     + (ddT / fmaxf(pc - 1.f, 1.f)) / 100.f;
    out[0] = loss;
  }
}

extern "C" void kernel_launch(void* const* d_in, const int* in_sizes, int n_in,
                              void* d_out, int out_size, void* d_ws, size_t ws_size,
                              hipStream_t stream) {
  const float* logits = (const float*)d_in[0];   // (S,B,D) f32
  const float* tgt    = (const float*)d_in[1];   // (S,B,D) f32
  const int*   maskp  = (const int*)d_in[2];     // (B,S) bool->int
  float* ws  = (float*)d_ws;                     // needs ~784 KB
  float* out = (float*)d_out;

  // 256 b x 4 s-chunks x 2 K-halves, one wave32 per block
  gram_pass1<<<dim3(B_DIM * 4 * 2), dim3(32), 0, stream>>>(logits, tgt, ws);
  finalize_pass2<<<dim3(1), dim3(256), 0, stream>>>(maskp, ws, out);
}